// SAXSLoss_48215302865276
// MI455X (gfx1250) — compile-verified
//
#include <hip/hip_runtime.h>
#include <hip/hip_bf16.h>

// ---------------------------------------------------------------------------
// SAXS P(r) L1 loss, MI455X (gfx1250, wave32).
//   Gram tiles via V_WMMA_F32_16X16X4_F32, two tiles (16x32 strip) per
//   iteration sharing the A operand; tile bookkeeping forced onto the SALU
//   via readfirstlane so EXEC stays all-ones around the WMMAs;
//   t = sqrt(4*d2) via raw v_sqrt_f32 + v_fract_f32 soft binning into
//   half-wave-private LDS histograms (ds_add_f32); block flush via
//   agent-scope fp atomics; tiny finalize kernel for the L1 sum.
// ---------------------------------------------------------------------------

#define NRES   256
#define NATOM  37
#define NPTS   (NRES * NATOM)        // 9472 atoms (== 16 * 592, exact tiles)
#define NT     (NPTS / 16)           // 592 tiles per dimension
#define NTH    (NT / 2)              // 296 column-tile pairs (strips)
#define NBINS  201
#define NBPAD  208                   // padded copy stride: rotates 16 LDS banks
#define TCLIP  ((float)NBINS - 1.000001f)

#define BLOCK  256
#define WPB    (BLOCK / 32)          // 8 waves per block (wave32)
#define HCOPIES (2 * WPB * 2)        // structures x waves x half-waves = 32
#define NBLOCKS 1024

// workspace layout (floats):
//   [0, 402)              : global histograms, 2 structures x 201 bins
//   [512, 512+NPTS)       : 4*|x|^2, structure 0 (pred)
//   [512+NPTS, +2*NPTS)   : 4*|x|^2, structure 1 (true)
#define WS_HIST 0
#define WS_SQ   512

typedef __attribute__((ext_vector_type(2))) float v2f;
typedef __attribute__((ext_vector_type(8))) float v8f;

// ---------------------------------------------------------------------------
// Kernel 1: zero global histograms + precompute 4*|x|^2 for both structures.
// ---------------------------------------------------------------------------
__global__ __launch_bounds__(BLOCK) void saxs_prep(const float* __restrict__ pred,
                                                   const float* __restrict__ truep,
                                                   float* __restrict__ ws) {
  if (blockIdx.x == 0) {
    for (int b = threadIdx.x; b < 2 * NBINS; b += BLOCK) ws[WS_HIST + b] = 0.0f;
  }
  int i = blockIdx.x * BLOCK + threadIdx.x;
  if (i < NPTS) {
    float x = pred[3 * i], y = pred[3 * i + 1], z = pred[3 * i + 2];
    ws[WS_SQ + i] = 4.0f * (x * x + y * y + z * z);
    x = truep[3 * i]; y = truep[3 * i + 1]; z = truep[3 * i + 2];
    ws[WS_SQ + NPTS + i] = 4.0f * (x * x + y * y + z * z);
  }
}

// ---------------------------------------------------------------------------
// Soft-bin one 16x16 Gram tile (8 accumulator elements per lane).
// DIAG=true only for the 592 diagonal tiles needing self-pair masking.
// ---------------------------------------------------------------------------
template <bool DIAG>
__device__ __forceinline__ void binTile(const v8f& C,
                                        const float* __restrict__ sqr4,
                                        const float* __restrict__ mr,
                                        float sqc4, float wm,
                                        int rbase, int cIdx,
                                        float* __restrict__ h) {
#pragma unroll
  for (int v = 0; v < 8; ++v) {
    float w = mr[v] * wm;
    if (DIAG) {
      if (rbase + v == cIdx) w = 0.0f;             // exclude self-pairs
    }
    // t = 2*sqrt(max(d2,1e-12)) == sqrt(max(4*d2, 4e-12)); 4*d2 via fma(-8).
    float d4 = __builtin_fmaf(C[v], -8.0f, sqr4[v] + sqc4);
    float tt = fminf(__builtin_amdgcn_sqrtf(fmaxf(d4, 4e-12f)), TCLIP);
    int   lo = (int)tt;                            // tt >= 0 -> trunc == floor
    float fr = __builtin_amdgcn_fractf(tt);        // v_fract_f32
    float wf = w * fr;
    __hip_atomic_fetch_add(&h[lo],     w - wf,
                           __ATOMIC_RELAXED, __HIP_MEMORY_SCOPE_WORKGROUP);
    __hip_atomic_fetch_add(&h[lo + 1], wf,         // lo <= 200 (TCLIP rounds to 200.0f); lo+1 stays in NBPAD
                           __ATOMIC_RELAXED, __HIP_MEMORY_SCOPE_WORKGROUP);
  }
}

// ---------------------------------------------------------------------------
// Kernel 2: persistent waves; each wave owns a 16x32 pair strip per iteration
// (two WMMAs sharing the A operand and all row-side data). All strip/tile
// bookkeeping is done on readfirstlane'd scalars so branches are s_cbranch
// and EXEC remains all-ones across the WMMAs.
// ---------------------------------------------------------------------------
__global__ __launch_bounds__(BLOCK) void saxs_hist(const float* __restrict__ pred,
                                                   const float* __restrict__ truep,
                                                   const float* __restrict__ mask,
                                                   float* __restrict__ ws) {
  __shared__ float lh[HCOPIES * NBPAD];            // 26.6 KB
  for (int i = threadIdx.x; i < HCOPIES * NBPAD; i += BLOCK) lh[i] = 0.0f;
  __syncthreads();

  const int lane = threadIdx.x & 31;
  const int half = lane >> 4;                      // 0: K=0,1  1: K=2,3
  const int lmod = lane & 15;
  // Wave-uniform scalars: force into SGPRs so all control flow is scalar.
  const int wib     = __builtin_amdgcn_readfirstlane(threadIdx.x >> 5);
  const int waveG   = __builtin_amdgcn_readfirstlane(blockIdx.x) * WPB + wib;
  const int waveTot = gridDim.x * WPB;
  const int TSH     = NT * NTH;                    // strips per structure
  const int TOTS    = 2 * TSH;                     // both structures
  const float* sq0  = ws + WS_SQ;

  for (int t = waveG; t < TOTS; t += waveTot) {    // t is SGPR-resident
    const int s   = t / TSH;
    const int r   = t - s * TSH;
    const int ti  = r / NTH;
    const int jp  = r - ti * NTH;
    const int tj0 = jp << 1;                       // strip: tiles tj0, tj0+1
    if (ti > tj0 + 1) continue;                    // scalar branch

    const float* pos = s ? truep : pred;
    const float* sq4 = sq0 + s * NPTS;             // holds 4*|x|^2
    float* h = lh + (((s * WPB + wib) << 1) | half) * NBPAD;  // half-wave copy

    const int i0 = ti << 4;
    const int j0 = tj0 << 4;                       // 32 columns j0..j0+31

    // ---- Row side (shared by both tiles): A operand + sq/mask octets. ----
    const float* rp = pos + 3 * (i0 + lmod);
    float rx = rp[0], ry = rp[1], rz = rp[2];
    v2f A = {half ? rz : rx, half ? 0.0f : ry};

    const int rbase = i0 + (half << 3);            // 32B aligned
    const float4 sqa = ((const float4*)(sq4 + rbase))[0];
    const float4 sqb = ((const float4*)(sq4 + rbase))[1];
    const float4 ma  = ((const float4*)(mask + rbase))[0];
    const float4 mb  = ((const float4*)(mask + rbase))[1];
    const float sqr4[8] = {sqa.x, sqa.y, sqa.z, sqa.w, sqb.x, sqb.y, sqb.z, sqb.w};
    const float mr[8]   = {ma.x,  ma.y,  ma.z,  ma.w,  mb.x,  mb.y,  mb.z,  mb.w};

    // ---- Column side: two B operands, two back-to-back WMMAs. ----
    const int cIdx0 = j0 + lmod;
    const int cIdx1 = cIdx0 + 16;
    const float* cp0 = pos + 3 * cIdx0;
    const float* cp1 = pos + 3 * cIdx1;
    float c0x = cp0[0], c0y = cp0[1], c0z = cp0[2];
    float c1x = cp1[0], c1y = cp1[1], c1z = cp1[2];
    v2f B0 = {half ? c0z : c0x, half ? 0.0f : c0y};
    v2f B1 = {half ? c1z : c1x, half ? 0.0f : c1y};

    const v8f Z = {0.f, 0.f, 0.f, 0.f, 0.f, 0.f, 0.f, 0.f};
    v8f C0 = __builtin_amdgcn_wmma_f32_16x16x4_f32(false, A, false, B0,
                                                   (short)0, Z, false, false);
    v8f C1 = __builtin_amdgcn_wmma_f32_16x16x4_f32(false, A, false, B1,
                                                   (short)0, Z, false, false);

    // ---- Epilogues (scalar branches; DIAG path is 0.3% of tiles). ----
    if (ti <= tj0) {
      const float sqc4 = sq4[cIdx0];
      const float mc   = mask[cIdx0];
      if (ti == tj0) binTile<true >(C0, sqr4, mr, sqc4, mc * 1.0f, rbase, cIdx0, h);
      else           binTile<false>(C0, sqr4, mr, sqc4, mc * 2.0f, rbase, cIdx0, h);
    }
    {   // ti <= tj0+1 guaranteed by the strip skip above
      const float sqc4 = sq4[cIdx1];
      const float mc   = mask[cIdx1];
      if (ti == tj0 + 1) binTile<true >(C1, sqr4, mr, sqc4, mc * 1.0f, rbase, cIdx1, h);
      else               binTile<false>(C1, sqr4, mr, sqc4, mc * 2.0f, rbase, cIdx1, h);
    }
  }

  // Reduce the 16 block-local copies per bin, flush once per block.
  __syncthreads();
  for (int idx = threadIdx.x; idx < 2 * NBINS; idx += BLOCK) {
    const int s = idx >= NBINS;
    const int b = idx - s * NBINS;
    float sum = 0.0f;
#pragma unroll
    for (int c = 0; c < 2 * WPB; ++c)
      sum += lh[(((s * WPB) << 1) + c) * NBPAD + b];
    __hip_atomic_fetch_add(&ws[WS_HIST + idx], sum,
                           __ATOMIC_RELAXED, __HIP_MEMORY_SCOPE_AGENT);
  }
}

// ---------------------------------------------------------------------------
// Kernel 3: normalize both histograms, L1-sum the difference.
// ---------------------------------------------------------------------------
__global__ __launch_bounds__(BLOCK) void saxs_final(const float* __restrict__ ws,
                                                    float* __restrict__ out) {
  __shared__ float red[BLOCK];
  float s0 = 0.0f, s1 = 0.0f;
  for (int b = threadIdx.x; b < NBINS; b += BLOCK) {
    s0 += ws[WS_HIST + b];
    s1 += ws[WS_HIST + NBINS + b];
  }
  red[threadIdx.x] = s0;
  __syncthreads();
  for (int off = BLOCK / 2; off > 0; off >>= 1) {
    if (threadIdx.x < off) red[threadIdx.x] += red[threadIdx.x + off];
    __syncthreads();
  }
  const float S0 = red[0];
  __syncthreads();
  red[threadIdx.x] = s1;
  __syncthreads();
  for (int off = BLOCK / 2; off > 0; off >>= 1) {
    if (threadIdx.x < off) red[threadIdx.x] += red[threadIdx.x + off];
    __syncthreads();
  }
  const float S1 = red[0];
  __syncthreads();

  const float inv0 = 1.0f / (S0 + 1e-12f);
  const float inv1 = 1.0f / (S1 + 1e-12f);
  float l1 = 0.0f;
  for (int b = threadIdx.x; b < NBINS; b += BLOCK)
    l1 += fabsf(ws[WS_HIST + b] * inv0 - ws[WS_HIST + NBINS + b] * inv1);
  red[threadIdx.x] = l1;
  __syncthreads();
  for (int off = BLOCK / 2; off > 0; off >>= 1) {
    if (threadIdx.x < off) red[threadIdx.x] += red[threadIdx.x + off];
    __syncthreads();
  }
  if (threadIdx.x == 0) out[0] = red[0];
}

// ---------------------------------------------------------------------------
extern "C" void kernel_launch(void* const* d_in, const int* in_sizes, int n_in,
                              void* d_out, int out_size, void* d_ws, size_t ws_size,
                              hipStream_t stream) {
  const float* pred  = (const float*)d_in[0];   // final_atom_positions
  const float* truep = (const float*)d_in[1];   // all_atom_positions
  const float* msk   = (const float*)d_in[2];   // all_atom_mask
  float* out = (float*)d_out;
  float* ws  = (float*)d_ws;

  saxs_prep<<<(NPTS + BLOCK - 1) / BLOCK, BLOCK, 0, stream>>>(pred, truep, ws);
  saxs_hist<<<NBLOCKS, BLOCK, 0, stream>>>(pred, truep, msk, ws);
  saxs_final<<<1, BLOCK, 0, stream>>>(ws, out);
}